// GraphAttentionLayer_65128884076838
// MI455X (gfx1250) — compile-verified
//
#include <hip/hip_runtime.h>
#include <hip/hip_bf16.h>
#include <stdint.h>

#define N_NODES 8192
#define IN_FEAT 256
#define OUT_FEAT 128
#define N_EDGES 262144
#define LRELU_ALPHA 0.2f
#define MASK_WORDS (N_NODES / 32)   // 256 u32 words per row
#define NEIGH_CAP 1024              // max degree ~60 expected; huge safety margin

typedef float v2f __attribute__((ext_vector_type(2)));
typedef float v8f __attribute__((ext_vector_type(8)));

// ---------------------------------------------------------------------------
// Kernel 1: Wh = h @ W^T + b_lin  via V_WMMA_F32_16X16X4_F32
// One wave computes a 16(M) x 64(N) strip = four 16x16 tiles, reusing the
// A fragment across the 4 N-tiles: 5 loads / 4 WMMAs per K-step.
// A (16x4, M x K) lane layout: lane L -> M = L%16, VGPR v -> K = 2*(L/16)+v
// B (4x16, K x N) mirrored:    lane L -> N = L%16, VGPR v -> K = 2*(L/16)+v
// C/D (16x16):                 VGPR r -> M = r + 8*(L/16), N = L%16
// ---------------------------------------------------------------------------
__global__ __launch_bounds__(32) void gat_wh_wmma(
    const float* __restrict__ h, const float* __restrict__ W,
    const float* __restrict__ b_lin, float* __restrict__ Wh) {
  const int m0   = blockIdx.x * 16;
  const int n0   = blockIdx.y * 64;   // 64-wide N strip
  const int lane = threadIdx.x;
  const int r    = lane & 15;
  const int half = lane >> 4;

  const float* __restrict__ hrow = h + (size_t)(m0 + r) * IN_FEAT + 2 * half;
  const float* __restrict__ wrow = W + (size_t)(n0 + r) * IN_FEAT + 2 * half;

  v8f c0 = {}, c1 = {}, c2 = {}, c3 = {};
  #pragma unroll 4
  for (int k = 0; k < IN_FEAT; k += 4) {
    v2f a  = *(const v2f*)(hrow + k);                  // shared A fragment
    v2f b0 = *(const v2f*)(wrow + k);                  // N-tile 0
    v2f b1 = *(const v2f*)(wrow + 16 * IN_FEAT + k);   // N-tile 1 (+16KB)
    v2f b2 = *(const v2f*)(wrow + 32 * IN_FEAT + k);   // N-tile 2 (+32KB)
    v2f b3 = *(const v2f*)(wrow + 48 * IN_FEAT + k);   // N-tile 3 (+48KB)
    // 8 args: (neg_a, A, neg_b, B, c_mod, C, reuse_a, reuse_b)
    c0 = __builtin_amdgcn_wmma_f32_16x16x4_f32(false, a, false, b0, (short)0,
                                               c0, false, false);
    c1 = __builtin_amdgcn_wmma_f32_16x16x4_f32(false, a, false, b1, (short)0,
                                               c1, false, false);
    c2 = __builtin_amdgcn_wmma_f32_16x16x4_f32(false, a, false, b2, (short)0,
                                               c2, false, false);
    c3 = __builtin_amdgcn_wmma_f32_16x16x4_f32(false, a, false, b3, (short)0,
                                               c3, false, false);
  }

  v8f acc[4] = {c0, c1, c2, c3};
  #pragma unroll
  for (int t = 0; t < 4; ++t) {
    const int col  = n0 + 16 * t + r;
    const float bias = b_lin[col];
    #pragma unroll
    for (int v = 0; v < 8; ++v) {
      const int row = m0 + v + 8 * half;
      Wh[(size_t)row * OUT_FEAT + col] = acc[t][v] + bias;
    }
  }
}

// ---------------------------------------------------------------------------
// Kernel 2: s1[i] = Wh[i]·a1, s2[i] = Wh[i]·a2. One wave32 per row,
// each lane owns one float4 (128 feats / 4 = 32 lanes), shuffle-reduce.
// ---------------------------------------------------------------------------
__global__ __launch_bounds__(256) void gat_scores(
    const float* __restrict__ Wh, const float* __restrict__ a,
    float* __restrict__ s1, float* __restrict__ s2) {
  const int wave = (blockIdx.x * blockDim.x + threadIdx.x) >> 5;
  const int lane = threadIdx.x & 31;
  if (wave >= N_NODES) return;

  const float4 w  = ((const float4*)(Wh + (size_t)wave * OUT_FEAT))[lane];
  const float4 x1 = ((const float4*)(a))[lane];
  const float4 x2 = ((const float4*)(a + OUT_FEAT))[lane];
  float p1 = w.x * x1.x + w.y * x1.y + w.z * x1.z + w.w * x1.w;
  float p2 = w.x * x2.x + w.y * x2.y + w.z * x2.z + w.w * x2.w;
  #pragma unroll
  for (int off = 16; off; off >>= 1) {
    p1 += __shfl_down(p1, off);
    p2 += __shfl_down(p2, off);
  }
  if (lane == 0) { s1[wave] = p1; s2[wave] = p2; }
}

// ---------------------------------------------------------------------------
// Kernel 3: clear adjacency bitmask (8 MB, L2-resident).
// ---------------------------------------------------------------------------
__global__ __launch_bounds__(256) void gat_clear(uint32_t* __restrict__ mask,
                                                 int nwords) {
  const int i = blockIdx.x * blockDim.x + threadIdx.x;
  if (i < nwords) mask[i] = 0u;
}

// ---------------------------------------------------------------------------
// Kernel 4: scatter edges into bitmask (atomicOr dedupes exactly like the
// dense adjacency .at[...].set(1.0) in the reference). edge_index is int64.
// ---------------------------------------------------------------------------
__global__ __launch_bounds__(256) void gat_scatter(
    const long long* __restrict__ ei, uint32_t* __restrict__ mask) {
  const int e = blockIdx.x * blockDim.x + threadIdx.x;
  if (e >= N_EDGES) return;
  const int src = (int)ei[e];
  const int dst = (int)ei[N_EDGES + e];
  atomicOr(&mask[(size_t)src * MASK_WORDS + (dst >> 5)], 1u << (dst & 31));
}

// ---------------------------------------------------------------------------
// Kernel 5: per-row masked softmax + aggregation + ELU.
// Block = 128 threads (4 waves) per row i. Phase 1: bit-scan mask words,
// build neighbor list + leaky-relu scores in LDS, block-max. Phase 2:
// exp(score-max), block-sum. Phase 3: thread t owns feature t; coalesced
// gather-accumulate of Wh[j][t] over neighbors.
// ---------------------------------------------------------------------------
__global__ __launch_bounds__(128) void gat_row(
    const float* __restrict__ Wh, const float* __restrict__ s1,
    const float* __restrict__ s2, const float* __restrict__ b_att,
    const uint32_t* __restrict__ mask, float* __restrict__ out) {
  __shared__ int   jl[NEIGH_CAP];
  __shared__ float wl[NEIGH_CAP];
  __shared__ int   cnt;
  __shared__ float red[128];

  const int i   = blockIdx.x;
  const int tid = threadIdx.x;
  if (tid == 0) cnt = 0;
  __syncthreads();

  const float s1i = s1[i];
  const float c0  = b_att[0];

  float lmax = -3.0e38f;
  for (int w = tid; w < MASK_WORDS; w += 128) {
    uint32_t m = mask[(size_t)i * MASK_WORDS + w];
    while (m) {
      const int b = __ffs((int)m) - 1;
      m &= m - 1u;
      const int j = (w << 5) + b;
      const float x  = s1i + s2[j] + c0;
      const float sc = (x >= 0.f) ? x : LRELU_ALPHA * x;   // leaky_relu(0.2)
      const int pos = atomicAdd(&cnt, 1);
      if (pos < NEIGH_CAP) { jl[pos] = j; wl[pos] = sc; }
      lmax = fmaxf(lmax, sc);
    }
  }
  red[tid] = lmax;
  __syncthreads();
  #pragma unroll
  for (int s = 64; s; s >>= 1) {
    if (tid < s) red[tid] = fmaxf(red[tid], red[tid + s]);
    __syncthreads();
  }
  const float mx = red[0];
  __syncthreads();

  const int n = min(cnt, NEIGH_CAP);
  float lsum = 0.f;
  for (int idx = tid; idx < n; idx += 128) {
    const float e = expf(wl[idx] - mx);
    wl[idx] = e;
    lsum += e;
  }
  red[tid] = lsum;
  __syncthreads();
  #pragma unroll
  for (int s = 64; s; s >>= 1) {
    if (tid < s) red[tid] += red[tid + s];
    __syncthreads();
  }
  const float denom = red[0];

  float acc = 0.f;
  if (n > 0) {
    for (int idx = 0; idx < n; ++idx)
      acc += wl[idx] * Wh[(size_t)jl[idx] * OUT_FEAT + tid];
    acc *= (1.f / denom);
  } else {
    // All-NEG_INF row: softmax is uniform 1/N over every node (exact match
    // to the dense reference). Essentially never taken (P ~ exp(-32)).
    for (int j = 0; j < N_NODES; ++j)
      acc += Wh[(size_t)j * OUT_FEAT + tid];
    acc *= (1.f / (float)N_NODES);
  }
  // elu (alpha=1)
  out[(size_t)i * OUT_FEAT + tid] = (acc > 0.f) ? acc : (expf(acc) - 1.f);
}

// ---------------------------------------------------------------------------
extern "C" void kernel_launch(void* const* d_in, const int* in_sizes, int n_in,
                              void* d_out, int out_size, void* d_ws,
                              size_t ws_size, hipStream_t stream) {
  const float*     h     = (const float*)d_in[0];
  const long long* ei    = (const long long*)d_in[1];   // int64 edge_index
  const float*     W     = (const float*)d_in[2];
  const float*     b_lin = (const float*)d_in[3];
  const float*     a     = (const float*)d_in[4];
  const float*     b_att = (const float*)d_in[5];
  float*           out   = (float*)d_out;

  char* ws = (char*)d_ws;
  float*    Wh   = (float*)ws;                                   // 4 MB
  float*    s1   = (float*)(ws + (size_t)N_NODES * OUT_FEAT * 4);
  float*    s2   = s1 + N_NODES;
  uint32_t* mask = (uint32_t*)(s2 + N_NODES);                    // 8 MB

  // 1. Wh = h @ W^T + b_lin  (WMMA f32; one wave per 16x64 strip)
  gat_wh_wmma<<<dim3(N_NODES / 16, OUT_FEAT / 64), 32, 0, stream>>>(h, W,
                                                                    b_lin, Wh);
  // 2. s1, s2
  gat_scores<<<(N_NODES * 32) / 256, 256, 0, stream>>>(Wh, a, s1, s2);
  // 3-4. adjacency bitmask
  gat_clear<<<(N_NODES * MASK_WORDS + 255) / 256, 256, 0, stream>>>(mask,
                                                       N_NODES * MASK_WORDS);
  gat_scatter<<<(N_EDGES + 255) / 256, 256, 0, stream>>>(ei, mask);
  // 5. masked softmax + aggregate + ELU
  gat_row<<<N_NODES, 128, 0, stream>>>(Wh, s1, s2, b_att, mask, out);
}